// TensorODEBLOCK_32641751450081
// MI455X (gfx1250) — compile-verified
//
#include <hip/hip_runtime.h>
#include <hip/hip_bf16.h>
#include <stddef.h>

typedef __attribute__((ext_vector_type(2))) float v2f;
typedef __attribute__((ext_vector_type(8))) float v8f;

#define NROW 2048
#define NCOL 512
#define NE   (NROW * NCOL)     // 1048576 elements per state buffer
#define LD   512
#define AS_STRIDE 36           // 144B rows: 16B aligned, bank-spread
#define BS_STRIDE 34           // even (b64 aligned), bank-spread

// Dormand-Prince error-estimate coefficients (e2 == 0)
__device__ constexpr float E1c = 71.0f / 57600.0f;
__device__ constexpr float E3c = -71.0f / 16695.0f;
__device__ constexpr float E4c = 71.0f / 1920.0f;
__device__ constexpr float E5c = -17253.0f / 339200.0f;
__device__ constexpr float E6c = 22.0f / 525.0f;
__device__ constexpr float E7c = -1.0f / 40.0f;

struct Ctrl {
    float t, h_abs, h, h0, d1save;
    int rejected, commit;
    float acc0, acc1, acc2;
};

// ---------------------------------------------------------------------------
// C[2048x512] = A[2048x512] * B[512x512]   (row-major fp32, v_wmma_f32_16x16x4)
// 256 threads = 8 waves; block tile 128x64; wave tile 32x32 (2x2 WMMA accums).
// Ping-pong LDS + register prefetch: one barrier per K-tile, global loads for
// tile kt+32 in flight while WMMAs consume tile kt. B tile stored transposed
// in LDS so both A and B fragments are contiguous ds_load_b64 (no v_mov packs).
// ---------------------------------------------------------------------------
__global__ __launch_bounds__(256) void gemm_wmma_f32(
    const float* __restrict__ A, const float* __restrict__ Bm,
    float* __restrict__ C) {
    __shared__ float As[2][128 * AS_STRIDE];   // [row][k]
    __shared__ float BsT[2][64 * BS_STRIDE];   // [col][k]  (transposed tile)

    const int tid     = threadIdx.x;
    const int rowBase = blockIdx.y * 128;
    const int colBase = blockIdx.x * 64;
    const int waveId  = tid >> 5;
    const int lane    = tid & 31;
    const int lh      = lane & 15;
    const int khalf   = (lane >> 4) << 1;   // 0 or 2
    const int waveRow = waveId >> 1;        // 0..3 -> 32-row slabs
    const int waveCol = waveId & 1;         // 0..1 -> 32-col slabs

    v8f acc00 = {}, acc01 = {}, acc10 = {}, acc11 = {};
    float4 aReg[4], bReg[2];

    auto fetch = [&](int kt) {
#pragma unroll
        for (int j = 0; j < 4; ++j) {            // A tile 128x32 = 1024 float4
            int f = tid + j * 256;
            int r = f >> 3, c = (f & 7) << 2;
            aReg[j] = *(const float4*)(A + (size_t)(rowBase + r) * LD + kt + c);
        }
#pragma unroll
        for (int j = 0; j < 2; ++j) {            // B tile 32x64 = 512 float4
            int f = tid + j * 256;
            int r = f >> 4, c = (f & 15) << 2;
            bReg[j] = *(const float4*)(Bm + (size_t)(kt + r) * LD + colBase + c);
        }
    };
    auto stage = [&](int p) {
#pragma unroll
        for (int j = 0; j < 4; ++j) {
            int f = tid + j * 256;
            int r = f >> 3, c = (f & 7) << 2;
            float* s = &As[p][r * AS_STRIDE + c];
            s[0] = aReg[j].x; s[1] = aReg[j].y; s[2] = aReg[j].z; s[3] = aReg[j].w;
        }
#pragma unroll
        for (int j = 0; j < 2; ++j) {            // transpose into [col][k]
            int f = tid + j * 256;
            int r = f >> 4, c = (f & 15) << 2;
            float* s = &BsT[p][c * BS_STRIDE + r];
            s[0 * BS_STRIDE] = bReg[j].x; s[1 * BS_STRIDE] = bReg[j].y;
            s[2 * BS_STRIDE] = bReg[j].z; s[3 * BS_STRIDE] = bReg[j].w;
        }
    };

    fetch(0);
    stage(0);
    __syncthreads();

    int p = 0;
    for (int kt = 0; kt < 512; kt += 32, p ^= 1) {
        const bool next = (kt + 32) < 512;
        if (next) fetch(kt + 32);                // global loads in flight

        const float* as = &As[p][(waveRow * 32 + lh) * AS_STRIDE + khalf];
        const float* bs = &BsT[p][(waveCol * 32 + lh) * BS_STRIDE + khalf];
#pragma unroll
        for (int k = 0; k < 32; k += 4) {
            v2f a0, a1, b0, b1;
            a0.x = as[k];                     a0.y = as[k + 1];
            a1.x = as[16 * AS_STRIDE + k];    a1.y = as[16 * AS_STRIDE + k + 1];
            b0.x = bs[k];                     b0.y = bs[k + 1];
            b1.x = bs[16 * BS_STRIDE + k];    b1.y = bs[16 * BS_STRIDE + k + 1];
            acc00 = __builtin_amdgcn_wmma_f32_16x16x4_f32(false, a0, false, b0, (short)0, acc00, false, false);
            acc01 = __builtin_amdgcn_wmma_f32_16x16x4_f32(false, a0, false, b1, (short)0, acc01, false, false);
            acc10 = __builtin_amdgcn_wmma_f32_16x16x4_f32(false, a1, false, b0, (short)0, acc10, false, false);
            acc11 = __builtin_amdgcn_wmma_f32_16x16x4_f32(false, a1, false, b1, (short)0, acc11, false, false);
        }

        if (next) stage(p ^ 1);                  // buf p^1 last read before prev barrier
        __syncthreads();
    }

    // C/D layout: VGPR m -> row m (lanes 0-15) / row m+8 (lanes 16-31), N = lane%16
    const int colOff = colBase + waveCol * 32 + lh;
    const int rowOff = rowBase + waveRow * 32 + (lane >> 4) * 8;
#pragma unroll
    for (int m = 0; m < 8; ++m) {
        C[(size_t)(rowOff + m) * LD + colOff]           = acc00[m];
        C[(size_t)(rowOff + m) * LD + colOff + 16]      = acc01[m];
        C[(size_t)(rowOff + 16 + m) * LD + colOff]      = acc10[m];
        C[(size_t)(rowOff + 16 + m) * LD + colOff + 16] = acc11[m];
    }
}

// MT[k][j] = M[j][k]
__global__ void transpose512(const float* __restrict__ M, float* __restrict__ MT) {
    int g = blockIdx.x * 256 + threadIdx.x;   // grid 1024 -> 262144
    int r = g >> 9, c = g & 511;
    MT[g] = M[c * 512 + r];
}

// out = y + h * (c1*p1 + c2*p2 + c3*p3 + c4*p4 + c5*p5); h read from *hp
__global__ __launch_bounds__(256) void combo_kernel(
    float* __restrict__ out, const float* __restrict__ y,
    const float* __restrict__ p1, float c1, const float* __restrict__ p2, float c2,
    const float* __restrict__ p3, float c3, const float* __restrict__ p4, float c4,
    const float* __restrict__ p5, float c5, const float* __restrict__ hp) {
    const int i = (blockIdx.x * 256 + threadIdx.x) * 4;
    const float h = *hp;
    float4 a = {0.f, 0.f, 0.f, 0.f};
    if (c1 != 0.f) { float4 v = *(const float4*)(p1 + i); a.x += c1*v.x; a.y += c1*v.y; a.z += c1*v.z; a.w += c1*v.w; }
    if (c2 != 0.f) { float4 v = *(const float4*)(p2 + i); a.x += c2*v.x; a.y += c2*v.y; a.z += c2*v.z; a.w += c2*v.w; }
    if (c3 != 0.f) { float4 v = *(const float4*)(p3 + i); a.x += c3*v.x; a.y += c3*v.y; a.z += c3*v.z; a.w += c3*v.w; }
    if (c4 != 0.f) { float4 v = *(const float4*)(p4 + i); a.x += c4*v.x; a.y += c4*v.y; a.z += c4*v.z; a.w += c4*v.w; }
    if (c5 != 0.f) { float4 v = *(const float4*)(p5 + i); a.x += c5*v.x; a.y += c5*v.y; a.z += c5*v.z; a.w += c5*v.w; }
    float4 yy = *(const float4*)(y + i);
    float4 o;
    o.x = yy.x + h * a.x; o.y = yy.y + h * a.y;
    o.z = yy.z + h * a.z; o.w = yy.w + h * a.w;
    *(float4*)(out + i) = o;
}

// acc0 += sum((h*err_i / scale_i)^2)
__global__ __launch_bounds__(256) void err_reduce_kernel(
    const float* __restrict__ y, const float* __restrict__ yn,
    const float* __restrict__ k1, const float* __restrict__ k3,
    const float* __restrict__ k4, const float* __restrict__ k5,
    const float* __restrict__ k6, const float* __restrict__ k7,
    Ctrl* __restrict__ ctrl) {
    __shared__ float red[256];
    const float h = ctrl->h;
    const int base = (blockIdx.x * 256 + threadIdx.x) * 4;
    float s = 0.f;
#pragma unroll
    for (int j = 0; j < 4; ++j) {
        int i = base + j;
        float e = h * (E1c*k1[i] + E3c*k3[i] + E4c*k4[i] + E5c*k5[i] + E6c*k6[i] + E7c*k7[i]);
        float sc = 1e-6f + 1e-3f * fmaxf(fabsf(y[i]), fabsf(yn[i]));
        float v = e / sc;
        s += v * v;
    }
    red[threadIdx.x] = s;
    __syncthreads();
    for (int off = 128; off > 0; off >>= 1) {
        if (threadIdx.x < off) red[threadIdx.x] += red[threadIdx.x + off];
        __syncthreads();
    }
    if (threadIdx.x == 0) atomicAdd(&ctrl->acc0, red[0]);
}

// acc0 += sum((y0/scale0)^2), acc1 += sum((f0/scale0)^2)
__global__ __launch_bounds__(256) void init_reduce0_kernel(
    const float* __restrict__ y0, const float* __restrict__ f0, Ctrl* __restrict__ ctrl) {
    __shared__ float r0[256], r1[256];
    const int base = (blockIdx.x * 256 + threadIdx.x) * 4;
    float s0 = 0.f, s1 = 0.f;
#pragma unroll
    for (int j = 0; j < 4; ++j) {
        int i = base + j;
        float yv = y0[i];
        float sc = 1e-6f + 1e-3f * fabsf(yv);
        float a = yv / sc, b = f0[i] / sc;
        s0 += a * a; s1 += b * b;
    }
    r0[threadIdx.x] = s0; r1[threadIdx.x] = s1;
    __syncthreads();
    for (int off = 128; off > 0; off >>= 1) {
        if (threadIdx.x < off) { r0[threadIdx.x] += r0[threadIdx.x + off]; r1[threadIdx.x] += r1[threadIdx.x + off]; }
        __syncthreads();
    }
    if (threadIdx.x == 0) { atomicAdd(&ctrl->acc0, r0[0]); atomicAdd(&ctrl->acc1, r1[0]); }
}

// acc2 += sum(((f1-f0)/scale0)^2)
__global__ __launch_bounds__(256) void init_reduce1_kernel(
    const float* __restrict__ y0, const float* __restrict__ f0,
    const float* __restrict__ f1, Ctrl* __restrict__ ctrl) {
    __shared__ float red[256];
    const int base = (blockIdx.x * 256 + threadIdx.x) * 4;
    float s = 0.f;
#pragma unroll
    for (int j = 0; j < 4; ++j) {
        int i = base + j;
        float sc = 1e-6f + 1e-3f * fabsf(y0[i]);
        float d = (f1[i] - f0[i]) / sc;
        s += d * d;
    }
    red[threadIdx.x] = s;
    __syncthreads();
    for (int off = 128; off > 0; off >>= 1) {
        if (threadIdx.x < off) red[threadIdx.x] += red[threadIdx.x + off];
        __syncthreads();
    }
    if (threadIdx.x == 0) atomicAdd(&ctrl->acc2, red[0]);
}

__global__ void ctrl_reset_kernel(Ctrl* c) {
    c->t = 0.f; c->h_abs = 0.f; c->h = 0.f; c->h0 = 0.f; c->d1save = 0.f;
    c->rejected = 0; c->commit = 0;
    c->acc0 = 0.f; c->acc1 = 0.f; c->acc2 = 0.f;
}

__global__ void ctrl_init_b_kernel(Ctrl* c) {
    float d0 = sqrtf(c->acc0 / (float)NE);
    float d1 = sqrtf(c->acc1 / (float)NE);
    float h0 = ((d0 < 1e-5f) || (d1 < 1e-5f)) ? 1e-6f : 0.01f * d0 / d1;
    c->h0 = h0; c->d1save = d1; c->acc2 = 0.f;
}

__global__ void ctrl_init_c_kernel(Ctrl* c) {
    float d2 = sqrtf(c->acc2 / (float)NE) / c->h0;
    float dmax = fmaxf(c->d1save, d2);
    float h1 = (dmax <= 1e-15f) ? fmaxf(1e-6f, c->h0 * 1e-3f) : powf(0.01f / dmax, 0.2f);
    float h_init = fminf(fminf(100.f * c->h0, h1), 1.0f);
    c->t = 0.f; c->h_abs = h_init; c->rejected = 0; c->commit = 0;
    c->h = fmaxf(fminf(h_init, 1.0f), 0.f);
    c->acc0 = 0.f;
}

__global__ void ctrl_step_kernel(Ctrl* c) {
    float err_norm = sqrtf(c->acc0 / (float)NE);
    float t = c->t, h = c->h;
    bool done   = (1.0f - t) <= 0.0f;
    bool accept = err_norm < 1.0f;
    float en  = (err_norm > 0.f) ? err_norm : 1.f;
    float fac = 0.9f * powf(en, -0.2f);
    float fac_acc = (err_norm == 0.f) ? 10.f : fminf(10.f, fac);
    if (c->rejected) fac_acc = fminf(fac_acc, 1.f);
    float fac_rej = fmaxf(0.2f, fac);
    if (!done) {
        if (accept) { c->t = t + h; c->h_abs = h * fac_acc; }
        else        { c->h_abs = h * fac_rej; }
        c->rejected = accept ? 0 : 1;
    }
    c->commit = (accept && !done) ? 1 : 0;
    c->h = fmaxf(fminf(c->h_abs, 1.0f - c->t), 0.f);
    c->acc0 = 0.f;
}

__global__ __launch_bounds__(256) void commit_kernel(
    float* __restrict__ y, const float* __restrict__ yn,
    float* __restrict__ fy, const float* __restrict__ k7,
    const Ctrl* __restrict__ c) {
    if (!c->commit) return;
    const int i = (blockIdx.x * 256 + threadIdx.x) * 4;
    *(float4*)(y + i)  = *(const float4*)(yn + i);
    *(float4*)(fy + i) = *(const float4*)(k7 + i);
}

extern "C" void kernel_launch(void* const* d_in, const int* in_sizes, int n_in,
                              void* d_out, int out_size, void* d_ws, size_t ws_size,
                              hipStream_t stream) {
    const float* x = (const float*)d_in[0];
    const float* P = (const float*)d_in[1];
    const float* M = (const float*)d_in[2];
    const float* F = (const float*)d_in[3];
    float* out = (float*)d_out;
    float* ws  = (float*)d_ws;

    float* y    = ws + (size_t)0 * NE;   // state A
    float* fy   = ws + (size_t)1 * NE;   // k1 = f(y)
    float* k2   = ws + (size_t)2 * NE;
    float* k3   = ws + (size_t)3 * NE;
    float* k4   = ws + (size_t)4 * NE;
    float* k5   = ws + (size_t)5 * NE;
    float* k6   = ws + (size_t)6 * NE;
    float* k7   = ws + (size_t)7 * NE;
    float* ytmp = ws + (size_t)8 * NE;
    float* ynew = ws + (size_t)9 * NE;
    float* MT   = ws + (size_t)10 * NE;  // 512*512
    Ctrl*  ctrl = (Ctrl*)(ws + (size_t)10 * NE + 512 * 512);
    float* hP   = (float*)((char*)ctrl + offsetof(Ctrl, h));
    float* h0P  = (float*)((char*)ctrl + offsetof(Ctrl, h0));

    // Butcher tableau (fp32, matching reference)
    const float A21 = 1.f/5.f;
    const float A31 = 3.f/40.f,  A32 = 9.f/40.f;
    const float A41 = 44.f/45.f, A42 = -56.f/15.f, A43 = 32.f/9.f;
    const float A51 = 19372.f/6561.f, A52 = -25360.f/2187.f, A53 = 64448.f/6561.f, A54 = -212.f/729.f;
    const float A61 = 9017.f/3168.f,  A62 = -355.f/33.f, A63 = 46732.f/5247.f, A64 = 49.f/176.f, A65 = -5103.f/18656.f;
    const float B1  = 35.f/384.f, B3 = 500.f/1113.f, B4 = 125.f/192.f, B5 = -2187.f/6784.f, B6 = 11.f/84.f;

    const dim3 gG(8, 16), bG(256);   // 512/64 cols x 2048/128 rows
    const int  RB = 1024;            // NE / (256*4)

    ctrl_reset_kernel<<<1, 1, 0, stream>>>(ctrl);
    transpose512<<<1024, 256, 0, stream>>>(M, MT);
    gemm_wmma_f32<<<gG, bG, 0, stream>>>(x, P, y);        // A0 = x @ P
    gemm_wmma_f32<<<gG, bG, 0, stream>>>(y, MT, fy);      // f0 = A0 @ M^T

    // scipy _select_initial_step
    init_reduce0_kernel<<<RB, 256, 0, stream>>>(y, fy, ctrl);
    ctrl_init_b_kernel<<<1, 1, 0, stream>>>(ctrl);
    combo_kernel<<<RB, 256, 0, stream>>>(ytmp, y, fy, 1.0f, fy, 0.f, fy, 0.f, fy, 0.f, fy, 0.f, h0P);
    gemm_wmma_f32<<<gG, bG, 0, stream>>>(ytmp, MT, k2);   // f1 (reuse k2 buffer)
    init_reduce1_kernel<<<RB, 256, 0, stream>>>(y, fy, k2, ctrl);
    ctrl_init_c_kernel<<<1, 1, 0, stream>>>(ctrl);

    for (int s = 0; s < 48; ++s) {
        combo_kernel<<<RB, 256, 0, stream>>>(ytmp, y, fy, A21, fy, 0.f, fy, 0.f, fy, 0.f, fy, 0.f, hP);
        gemm_wmma_f32<<<gG, bG, 0, stream>>>(ytmp, MT, k2);
        combo_kernel<<<RB, 256, 0, stream>>>(ytmp, y, fy, A31, k2, A32, fy, 0.f, fy, 0.f, fy, 0.f, hP);
        gemm_wmma_f32<<<gG, bG, 0, stream>>>(ytmp, MT, k3);
        combo_kernel<<<RB, 256, 0, stream>>>(ytmp, y, fy, A41, k2, A42, k3, A43, fy, 0.f, fy, 0.f, hP);
        gemm_wmma_f32<<<gG, bG, 0, stream>>>(ytmp, MT, k4);
        combo_kernel<<<RB, 256, 0, stream>>>(ytmp, y, fy, A51, k2, A52, k3, A53, k4, A54, fy, 0.f, hP);
        gemm_wmma_f32<<<gG, bG, 0, stream>>>(ytmp, MT, k5);
        combo_kernel<<<RB, 256, 0, stream>>>(ytmp, y, fy, A61, k2, A62, k3, A63, k4, A64, k5, A65, hP);
        gemm_wmma_f32<<<gG, bG, 0, stream>>>(ytmp, MT, k6);
        combo_kernel<<<RB, 256, 0, stream>>>(ynew, y, fy, B1, k3, B3, k4, B4, k5, B5, k6, B6, hP);
        gemm_wmma_f32<<<gG, bG, 0, stream>>>(ynew, MT, k7);
        err_reduce_kernel<<<RB, 256, 0, stream>>>(y, ynew, fy, k3, k4, k5, k6, k7, ctrl);
        ctrl_step_kernel<<<1, 1, 0, stream>>>(ctrl);
        commit_kernel<<<RB, 256, 0, stream>>>(y, ynew, fy, k7, ctrl);
    }

    gemm_wmma_f32<<<gG, bG, 0, stream>>>(y, F, out);      // A_f @ F
}